// GATv2Conv_Decoder_85950885527597
// MI455X (gfx1250) — compile-verified
//
#include <hip/hip_runtime.h>
#include <hip/hip_bf16.h>

// ---------------- CDNA5 WMMA types ----------------
typedef __attribute__((ext_vector_type(2))) float v2f;
typedef __attribute__((ext_vector_type(8))) float v8f;

#define HEADS 5
#define KDIM 64

// Monotone order-preserving float<->uint encoding for atomic max
__device__ __forceinline__ unsigned encF(float f) {
    unsigned u = __float_as_uint(f);
    return (u & 0x80000000u) ? ~u : (u | 0x80000000u);
}
__device__ __forceinline__ float decF(unsigned u) {
    return __uint_as_float((u & 0x80000000u) ? (u ^ 0x80000000u) : ~u);
}

__device__ __forceinline__ void atomicAddF(float* p, float v) {
    unsafeAtomicAdd(p, v);   // global_atomic_add_f32 on gfx1250
}

// ---------------- GEMM: out[M,LDN] = A[M,64] @ W[64,LDN] + bias[LDN] ----------------
// One wave -> 16 rows x (16*NTILES) cols via V_WMMA_F32_16X16X4_F32 (full fp32).
// LDN is a compile-time constant so all W/out offsets fold into the 24-bit
// instruction offset field (no per-load 64-bit address math).
template <int LDN, int NTILES>
__global__ __launch_bounds__(128) void gemm_wmma_f32(
    const float* __restrict__ A, const float* __restrict__ W,
    const float* __restrict__ bias, float* __restrict__ out, int M)
{
    const int lane = threadIdx.x & 31;
    const int wave = threadIdx.x >> 5;
    const int mt   = blockIdx.x * 4 + wave;
    const int rowBase = mt * 16;
    if (rowBase >= M) return;                 // wave-uniform exit
    const int colBase = blockIdx.y * (16 * NTILES);
    const int half = lane >> 4;               // 0 or 1
    const int lm   = lane & 15;

    int rowA = rowBase + lm;
    if (rowA >= M) rowA = M - 1;              // clamp (discarded at store)
    // A: per-lane base, constant offsets kk
    const float* __restrict__ abase = A + (size_t)rowA * KDIM + 2 * half;
    // W: per-lane base, constant offsets kk*LDN + nt*16
    const float* __restrict__ wbase = W + (size_t)(2 * half) * LDN + colBase + lm;

    v8f acc[NTILES];
#pragma unroll
    for (int nt = 0; nt < NTILES; ++nt) acc[nt] = v8f{};

#pragma unroll
    for (int kk = 0; kk < KDIM; kk += 4) {
        float2 av = *reinterpret_cast<const float2*>(abase + kk);
        v2f a; a.x = av.x; a.y = av.y;
#pragma unroll
        for (int nt = 0; nt < NTILES; ++nt) {
            v2f b;
            b.x = wbase[kk * LDN       + nt * 16];
            b.y = wbase[(kk + 1) * LDN + nt * 16];
            acc[nt] = __builtin_amdgcn_wmma_f32_16x16x4_f32(
                          false, a, false, b, (short)0, acc[nt], false, false);
        }
    }

    float* __restrict__ obase = out + (size_t)(rowBase + 8 * half) * LDN + colBase + lm;
    const bool inb = (rowBase + 8 * half + 7) < M;   // whole 8-row group in-bounds?
#pragma unroll
    for (int nt = 0; nt < NTILES; ++nt) {
        const float bcol = bias[colBase + nt * 16 + lm];
        if (inb) {
#pragma unroll
            for (int v = 0; v < 8; ++v)
                obase[v * LDN + nt * 16] = acc[nt][v] + bcol;
        } else {
#pragma unroll
            for (int v = 0; v < 8; ++v)
                if (rowBase + 8 * half + v < M)
                    obase[v * LDN + nt * 16] = acc[nt][v] + bcol;
        }
    }
}

// ---------------- Edge pass A: per-edge attention logits + segment max ----------------
// One wave per edge; lane covers 2 channels.
__global__ __launch_bounds__(256) void edge_alpha_kernel(
    const int* __restrict__ ei, int E0, int Etot,
    const float* __restrict__ xl, const float* __restrict__ xr,
    const float* __restrict__ att,
    float* __restrict__ alpha, unsigned* __restrict__ menc)
{
    const int lane = threadIdx.x & 31;
    const int e = blockIdx.x * 8 + (threadIdx.x >> 5);
    if (e >= Etot) return;
    const int src = (e < E0) ? ei[e]      : (e - E0);
    const int dst = (e < E0) ? ei[E0 + e] : (e - E0);
    const float* __restrict__ xls = xl + (size_t)src * (HEADS * KDIM);
    const float* __restrict__ xrd = xr + (size_t)dst * (HEADS * KDIM);
    const int c = 2 * lane;

#pragma unroll
    for (int h = 0; h < HEADS; ++h) {
        float2 u = *reinterpret_cast<const float2*>(xls + h * KDIM + c);
        float2 v = *reinterpret_cast<const float2*>(xrd + h * KDIM + c);
        float z0 = u.x + v.x, z1 = u.y + v.y;
        z0 = fmaxf(z0, 0.2f * z0);            // leaky_relu(z, 0.2)
        z1 = fmaxf(z1, 0.2f * z1);
        float2 aw = *reinterpret_cast<const float2*>(att + h * KDIM + c);
        float s = z0 * aw.x + z1 * aw.y;
#pragma unroll
        for (int off = 16; off; off >>= 1) s += __shfl_xor(s, off, 32);
        if (lane == 0) {
            alpha[(size_t)e * HEADS + h] = s;
            atomicMax(&menc[(size_t)dst * HEADS + h], encF(s));
        }
    }
}

// ---------------- Edge pass B: exp(alpha - max) + segment sum ----------------
__global__ __launch_bounds__(256) void edge_exp_kernel(
    const int* __restrict__ ei, int E0, int Etot,
    const float* __restrict__ alpha, const unsigned* __restrict__ menc,
    float* __restrict__ aexp, float* __restrict__ den)
{
    const int tid = blockIdx.x * 256 + threadIdx.x;
    if (tid >= Etot * HEADS) return;
    const int e = tid / HEADS;
    const int h = tid - e * HEADS;
    const int dst = (e < E0) ? ei[E0 + e] : (e - E0);
    const float m = decF(menc[(size_t)dst * HEADS + h]);
    const float a = __expf(alpha[tid] - m);
    aexp[tid] = a;
    atomicAddF(&den[(size_t)dst * HEADS + h], a);
}

// ---------------- Edge pass C: normalized weighted scatter, fused head-mean ----------------
// One wave per edge; lane covers 2 channels; mean over heads folded in (x0.2).
__global__ __launch_bounds__(256) void edge_agg_kernel(
    const int* __restrict__ ei, int E0, int Etot,
    const float* __restrict__ xl, const float* __restrict__ aexp,
    const float* __restrict__ den, float* __restrict__ agg)
{
    const int lane = threadIdx.x & 31;
    const int e = blockIdx.x * 8 + (threadIdx.x >> 5);
    if (e >= Etot) return;
    const int src = (e < E0) ? ei[e]      : (e - E0);
    const int dst = (e < E0) ? ei[E0 + e] : (e - E0);
    const float* __restrict__ xls = xl + (size_t)src * (HEADS * KDIM);

    float w[HEADS];
#pragma unroll
    for (int h = 0; h < HEADS; ++h)
        w[h] = aexp[(size_t)e * HEADS + h] /
               (den[(size_t)dst * HEADS + h] + 1e-16f);

    const int c = 2 * lane;
    float a0 = 0.f, a1 = 0.f;
#pragma unroll
    for (int h = 0; h < HEADS; ++h) {
        float2 u = *reinterpret_cast<const float2*>(xls + h * KDIM + c);
        a0 += w[h] * u.x;
        a1 += w[h] * u.y;
    }
    atomicAddF(&agg[(size_t)dst * KDIM + c],     0.2f * a0);  // /HEADS
    atomicAddF(&agg[(size_t)dst * KDIM + c + 1], 0.2f * a1);
}

// ---------------- out = relu(agg + bias) ----------------
__global__ __launch_bounds__(256) void finalize_kernel(
    const float* __restrict__ agg, const float* __restrict__ bias,
    float* __restrict__ out, int total)
{
    const int tid = blockIdx.x * 256 + threadIdx.x;
    if (tid >= total) return;
    out[tid] = fmaxf(agg[tid] + bias[tid & (KDIM - 1)], 0.f);
}

extern "C" void kernel_launch(void* const* d_in, const int* in_sizes, int n_in,
                              void* d_out, int out_size, void* d_ws, size_t ws_size,
                              hipStream_t stream) {
    const float* x    = (const float*)d_in[0];
    const int*   ei   = (const int*)  d_in[1];
    const float* W1l  = (const float*)d_in[2];
    const float* b1l  = (const float*)d_in[3];
    const float* W1r  = (const float*)d_in[4];
    const float* b1r  = (const float*)d_in[5];
    const float* att1 = (const float*)d_in[6];
    const float* bias1= (const float*)d_in[7];
    const float* W2l  = (const float*)d_in[8];
    const float* b2l  = (const float*)d_in[9];
    const float* W2r  = (const float*)d_in[10];
    const float* b2r  = (const float*)d_in[11];
    const float* att2 = (const float*)d_in[12];
    const float* bias2= (const float*)d_in[13];
    const float* Wfc  = (const float*)d_in[14];
    const float* bfc  = (const float*)d_in[15];

    const int N    = in_sizes[0] / KDIM;      // 50000
    const int E0   = in_sizes[1] / 2;         // 400000
    const int Etot = E0 + N;                  // +self loops
    const int HC   = HEADS * KDIM;            // 320

    // Workspace carving (deterministic)
    char* base = (char*)d_ws;
    size_t off = 0;
    auto carve = [&](size_t bytes) {
        char* p = base + off;
        off += (bytes + 255) & ~(size_t)255;
        return p;
    };
    float*    XL    = (float*)   carve((size_t)N * HC * 4);
    float*    XR    = (float*)   carve((size_t)N * HC * 4);
    float*    ALPHA = (float*)   carve((size_t)Etot * HEADS * 4);
    float*    AEXP  = (float*)   carve((size_t)Etot * HEADS * 4);
    unsigned* MENC  = (unsigned*)carve((size_t)N * HEADS * 4);
    float*    DEN   = (float*)   carve((size_t)N * HEADS * 4);
    float*    AGG   = (float*)   carve((size_t)N * KDIM * 4);
    float*    H     = (float*)   carve((size_t)N * KDIM * 4);
    (void)ws_size; (void)n_in; (void)out_size;

    const int Mtiles = (N + 15) / 16;
    dim3 gemmGrid((Mtiles + 3) / 4, HC / 64);    // 5 col-groups of 64
    dim3 fcGrid  ((Mtiles + 3) / 4, KDIM / 64);  // 1 col-group of 64
    const int edgeBlocks = (Etot + 7) / 8;
    const int expBlocks  = (Etot * HEADS + 255) / 256;
    const int finBlocks  = (N * KDIM + 255) / 256;

    // ---------------- Layer 1 ----------------
    hipMemsetAsync(MENC, 0, (size_t)N * HEADS * 4, stream);
    hipMemsetAsync(DEN,  0, (size_t)N * HEADS * 4, stream);
    hipMemsetAsync(AGG,  0, (size_t)N * KDIM  * 4, stream);
    gemm_wmma_f32<320, 4><<<gemmGrid, 128, 0, stream>>>(x, W1l, b1l, XL, N);
    gemm_wmma_f32<320, 4><<<gemmGrid, 128, 0, stream>>>(x, W1r, b1r, XR, N);
    edge_alpha_kernel<<<edgeBlocks, 256, 0, stream>>>(ei, E0, Etot, XL, XR, att1, ALPHA, MENC);
    edge_exp_kernel<<<expBlocks, 256, 0, stream>>>(ei, E0, Etot, ALPHA, MENC, AEXP, DEN);
    edge_agg_kernel<<<edgeBlocks, 256, 0, stream>>>(ei, E0, Etot, XL, AEXP, DEN, AGG);
    finalize_kernel<<<finBlocks, 256, 0, stream>>>(AGG, bias1, H, N * KDIM);

    // ---------------- Layer 2 ----------------
    hipMemsetAsync(MENC, 0, (size_t)N * HEADS * 4, stream);
    hipMemsetAsync(DEN,  0, (size_t)N * HEADS * 4, stream);
    hipMemsetAsync(AGG,  0, (size_t)N * KDIM  * 4, stream);
    gemm_wmma_f32<320, 4><<<gemmGrid, 128, 0, stream>>>(H, W2l, b2l, XL, N);
    gemm_wmma_f32<320, 4><<<gemmGrid, 128, 0, stream>>>(H, W2r, b2r, XR, N);
    edge_alpha_kernel<<<edgeBlocks, 256, 0, stream>>>(ei, E0, Etot, XL, XR, att2, ALPHA, MENC);
    edge_exp_kernel<<<expBlocks, 256, 0, stream>>>(ei, E0, Etot, ALPHA, MENC, AEXP, DEN);
    edge_agg_kernel<<<edgeBlocks, 256, 0, stream>>>(ei, E0, Etot, XL, AEXP, DEN, AGG);
    finalize_kernel<<<finBlocks, 256, 0, stream>>>(AGG, bias2, H, N * KDIM);

    // ---------------- Final FC ----------------
    gemm_wmma_f32<64, 4><<<fcGrid, 128, 0, stream>>>(H, Wfc, bfc, (float*)d_out, N);
}